// CausalConvolution_10694468567161
// MI455X (gfx1250) — compile-verified
//
#include <hip/hip_runtime.h>

typedef float v2f __attribute__((ext_vector_type(2)));
typedef float v8f __attribute__((ext_vector_type(8)));

#define NFFT   131072          // 2*T
#define TLEN   65536
#define ROWS   512             // B*C = 8*64
#define TWO_PI 6.28318530717958647692f
#define TILE_F2 272            // 256 complex + padding (one slot per 16)

__device__ __forceinline__ int pidx(int i) { return i + (i >> 4); }

__device__ __forceinline__ float2 cmulf(float2 a, float2 b) {
    return make_float2(a.x * b.x - a.y * b.y, a.x * b.y + a.y * b.x);
}
__device__ __forceinline__ float2 caddf(float2 a, float2 b) { return make_float2(a.x + b.x, a.y + b.y); }
__device__ __forceinline__ float2 csubf(float2 a, float2 b) { return make_float2(a.x - b.x, a.y - b.y); }
__device__ __forceinline__ float2 cisf(float ang) {
    float s, c; __sincosf(ang, &s, &c); return make_float2(c, s);
}

// A-operand register images of the (symmetric) 16-pt DFT matrix F16.
// Per ISA 7.12.2, f32 A 16x4: lanes 0-15 hold row M, K=4s+{0,1} in v0/v1;
// lanes 16-31 hold row M, K=4s+{2,3}.
struct Fregs { v2f fr[4]; v2f fi[4]; v2f fn[4]; };

__device__ __forceinline__ Fregs load_fregs(const float2* Ftab, int lane) {
    Fregs R;
    int r   = lane & 15;
    int hi2 = ((lane >> 4) & 1) << 1;          // 0 or 2
#pragma unroll
    for (int s = 0; s < 4; ++s) {
        int c0 = 4 * s + hi2;
        float2 a = Ftab[r * 16 + c0];
        float2 b = Ftab[r * 16 + c0 + 1];
        R.fr[s] = (v2f){a.x, b.x};
        R.fi[s] = (v2f){a.y, b.y};
        R.fn[s] = (v2f){-a.y, -b.y};
    }
    return R;
}

// B-operand (4x16) from LDS tile holding a natural-order length-256 complex
// array: element B[row=K][col] lives at natural index col + 16*row.
// Guessed B layout mirrors C/D: VGPR v holds K=4s+v (lanes 0-15) / 4s+v+2 (16-31).
__device__ __forceinline__ void loadB(const float2* tile, int lane, v2f Br[4], v2f Bi[4]) {
    int col = lane & 15;
    int hi2 = ((lane >> 4) & 1) << 1;
#pragma unroll
    for (int s = 0; s < 4; ++s) {
        int row0 = 4 * s + hi2;
        float2 c0 = tile[pidx(col + 16 * row0)];
        float2 c1 = tile[pidx(col + 16 * row0 + 16)];
        Br[s] = (v2f){c0.x, c1.x};
        Bi[s] = (v2f){c0.y, c1.y};
    }
}

// Complex 16x16x16 matmul D = F16 * B (INV => conj(F16)), via 16 chained
// v_wmma_f32_16x16x4_f32 (f32 WMMA has no A/B NEG => keep -Fi registers).
template <bool INV>
__device__ __forceinline__ void cmatmul(const Fregs& F, const v2f Br[4], const v2f Bi[4],
                                        v8f& Dr, v8f& Di) {
    v8f cr = {0.f, 0.f, 0.f, 0.f, 0.f, 0.f, 0.f, 0.f};
    v8f ci = {0.f, 0.f, 0.f, 0.f, 0.f, 0.f, 0.f, 0.f};
#pragma unroll
    for (int s = 0; s < 4; ++s)
        cr = __builtin_amdgcn_wmma_f32_16x16x4_f32(false, F.fr[s], false, Br[s], (short)0, cr, false, false);
#pragma unroll
    for (int s = 0; s < 4; ++s) {
        v2f fx = INV ? F.fi[s] : F.fn[s];       // Dr += (-Fi or +Fi) * Bi
        cr = __builtin_amdgcn_wmma_f32_16x16x4_f32(false, fx, false, Bi[s], (short)0, cr, false, false);
    }
#pragma unroll
    for (int s = 0; s < 4; ++s)
        ci = __builtin_amdgcn_wmma_f32_16x16x4_f32(false, F.fr[s], false, Bi[s], (short)0, ci, false, false);
#pragma unroll
    for (int s = 0; s < 4; ++s) {
        v2f fy = INV ? F.fn[s] : F.fi[s];       // Di += (+Fi or -Fi) * Br
        ci = __builtin_amdgcn_wmma_f32_16x16x4_f32(false, fy, false, Br[s], (short)0, ci, false, false);
    }
    Dr = cr; Di = ci;
}

// In-place 256-pt complex FFT (natural order in/out) on one wave's LDS tile.
// m = m1 + 16*m2, j = j2 + 16*j1:
//   stage A: A[m1,j2] = sum_m2 F16[j2,m2] z[m1+16m2]   (D rows j2, cols m1)
//            store T = A * W256^{m1*j2} at natural idx j2 + 16*m1
//   stage B: Z[j2+16j1] = sum_m1 F16[j1,m1] T[m1,j2]   (D rows j1, cols j2)
template <bool INV>
__device__ void fft256(float2* tile, int lane, const Fregs& F) {
    const float ds = INV ? 1.0f : -1.0f;
    v2f Br[4], Bi[4];
    v8f Dr, Di;
    int Ncol  = lane & 15;
    int Mbase = ((lane >> 4) & 1) << 3;

    loadB(tile, lane, Br, Bi);
    cmatmul<INV>(F, Br, Bi, Dr, Di);
#pragma unroll
    for (int v = 0; v < 8; ++v) {
        int j2 = Mbase + v, m1 = Ncol;
        float2 d = make_float2(Dr[v], Di[v]);
        d = cmulf(d, cisf(ds * TWO_PI * (float)((j2 * m1) & 255) * (1.0f / 256.0f)));
        tile[pidx(j2 + 16 * m1)] = d;           // transposed store + twiddle
    }

    loadB(tile, lane, Br, Bi);
    cmatmul<INV>(F, Br, Bi, Dr, Di);
#pragma unroll
    for (int v = 0; v < 8; ++v) {
        int j1 = Mbase + v;
        tile[pidx(Ncol + 16 * j1)] = make_float2(Dr[v], Di[v]);  // natural store
    }
}

__device__ __forceinline__ void build_ftab(float2* Ftab, int tid) {
    if (tid < 256) {
        int rr = tid >> 4, cc = tid & 15;
        Ftab[tid] = cisf(-TWO_PI * (float)((rr * cc) & 15) * (1.0f / 16.0f));
    }
}

// ---- Kernel 1: inner 256-FFT over n2 (stride 512) + twiddle W_N^{n1*k2} ----
__global__ void k_fwd_inner(const float* __restrict__ in, float2* __restrict__ W, int quant) {
    extern __shared__ char smem[];
    float2* Ftab  = (float2*)smem;              // 256
    float2* tiles = Ftab + 256;                 // 8 * TILE_F2
    int tid = threadIdx.x, lane = tid & 31, w = tid >> 5;
    build_ftab(Ftab, tid);
    __syncthreads();
    Fregs F = load_fregs(Ftab, lane);

    float2* tile = tiles + w * TILE_F2;
    int n1 = blockIdx.x * 8 + w;
    int r  = blockIdx.y;
    const float* rowp = in + (size_t)r * TLEN;

    for (int i = lane; i < 256; i += 32) {      // i = n2; t = n1 + 512*n2, zero-pad t>=T
        int t = n1 + 512 * i;
        float val = 0.f;
        if (t < TLEN) {
            val = rowp[t];
            if (quant) {                        // bf16 round-to-nearest-even
                unsigned u = __float_as_uint(val);
                u = (u + 0x7FFFu + ((u >> 16) & 1u)) & 0xFFFF0000u;
                val = __uint_as_float(u);
            }
        }
        tile[pidx(i)] = make_float2(val, 0.f);
    }
    fft256<false>(tile, lane, F);

    float2* outp = W + (size_t)r * NFFT + (size_t)n1 * 256;
    for (int i = lane; i < 256; i += 32) {      // i = k2
        float ang = -TWO_PI * (float)((n1 * i) & (NFFT - 1)) * (1.0f / (float)NFFT);
        outp[i] = cmulf(tile[pidx(i)], cisf(ang));
    }
}

// ---- Kernel 2: outer 512-FFT over n1, X*H*(1/N), inverse, conj twiddle ----
__global__ void k_outer_mul(float2* __restrict__ Wx, const float2* __restrict__ Wh) {
    extern __shared__ char smem[];
    float2* Ftab  = (float2*)smem;              // 256
    float2* Xt    = Ftab + 256;                 // [512][9] padded
    float2* Ht    = Xt + 512 * 9;
    float2* tiles = Ht + 512 * 9;               // 8 * TILE_F2
    int tid = threadIdx.x, lane = tid & 31, w = tid >> 5;
    build_ftab(Ftab, tid);

    int r = blockIdx.y;
    int k2base = blockIdx.x * 8;
    const float2* gx = Wx + (size_t)r * NFFT;
    const float2* gh = Wh + (size_t)r * NFFT;
    for (int idx = tid; idx < 4096; idx += 256) {      // cooperative coalesced load
        int n1 = idx >> 3, j = idx & 7;
        Xt[n1 * 9 + j] = gx[n1 * 256 + k2base + j];
        Ht[n1 * 9 + j] = gh[n1 * 256 + k2base + j];
    }
    __syncthreads();

    Fregs F = load_fregs(Ftab, lane);
    float2* tile = tiles + w * TILE_F2;
    int j = w, k2 = k2base + w;                 // each wave owns one k2 column

    for (int which = 0; which < 2; ++which) {   // forward 512-FFT on X then H
        float2* ct = which ? Ht : Xt;
        for (int u = lane; u < 256; u += 32) {  // radix-2 + W512^u
            float2 a = ct[u * 9 + j], b = ct[(u + 256) * 9 + j];
            float2 s0 = caddf(a, b);
            float2 s1 = cmulf(csubf(a, b), cisf(-TWO_PI * (float)u * (1.0f / 512.0f)));
            ct[u * 9 + j] = s0; ct[(u + 256) * 9 + j] = s1;
        }
        for (int half = 0; half < 2; ++half) {
            for (int u = lane; u < 256; u += 32) tile[pidx(u)] = ct[(u + half * 256) * 9 + j];
            fft256<false>(tile, lane, F);
            for (int u = lane; u < 256; u += 32) ct[(u + half * 256) * 9 + j] = tile[pidx(u)];
        }
    }

    for (int i = lane; i < 512; i += 32) {      // pointwise product, full 1/N scale
        float2 p = cmulf(Xt[i * 9 + j], Ht[i * 9 + j]);
        Xt[i * 9 + j] = make_float2(p.x * (1.0f / (float)NFFT), p.y * (1.0f / (float)NFFT));
    }

    for (int half = 0; half < 2; ++half) {      // inverse 256-FFT per half
        for (int u = lane; u < 256; u += 32) tile[pidx(u)] = Xt[(u + half * 256) * 9 + j];
        fft256<true>(tile, lane, F);
        for (int u = lane; u < 256; u += 32) Xt[(u + half * 256) * 9 + j] = tile[pidx(u)];
    }
    for (int u = lane; u < 256; u += 32) {      // radix-2 combine + conj W_N^{n1*k2}
        float2 q0 = Xt[u * 9 + j];
        float2 q1 = cmulf(Xt[(u + 256) * 9 + j], cisf(TWO_PI * (float)u * (1.0f / 512.0f)));
        float2 z0 = caddf(q0, q1), z1 = csubf(q0, q1);
        z0 = cmulf(z0, cisf(TWO_PI * (float)((u * k2) & (NFFT - 1)) * (1.0f / (float)NFFT)));
        z1 = cmulf(z1, cisf(TWO_PI * (float)(((u + 256) * k2) & (NFFT - 1)) * (1.0f / (float)NFFT)));
        Xt[u * 9 + j] = z0; Xt[(u + 256) * 9 + j] = z1;
    }
    __syncthreads();

    float2* gxo = Wx + (size_t)r * NFFT;
    for (int idx = tid; idx < 4096; idx += 256) {      // cooperative coalesced store
        int n1 = idx >> 3, jj = idx & 7;
        gxo[n1 * 256 + k2base + jj] = Xt[n1 * 9 + jj];
    }
}

// ---- Kernel 3: inverse inner 256-FFT over k2; emit y[t], t = n1 + 512*n2 < T ----
__global__ void k_inv_inner(const float2* __restrict__ Wz, float* __restrict__ out) {
    extern __shared__ char smem[];
    float2* Ftab  = (float2*)smem;
    float2* tiles = Ftab + 256;
    int tid = threadIdx.x, lane = tid & 31, w = tid >> 5;
    build_ftab(Ftab, tid);
    __syncthreads();
    Fregs F = load_fregs(Ftab, lane);

    float2* tile = tiles + w * TILE_F2;
    int n1 = blockIdx.x * 8 + w;
    int r  = blockIdx.y;
    const float2* inp = Wz + (size_t)r * NFFT + (size_t)n1 * 256;
    for (int i = lane; i < 256; i += 32) tile[pidx(i)] = inp[i];
    fft256<true>(tile, lane, F);

    float* outp = out + (size_t)r * TLEN;
    for (int i = lane; i < 128; i += 32) {      // n2 < 128 <=> t < T
        outp[n1 + 512 * i] = tile[pidx(i)].x;
    }
}

extern "C" void kernel_launch(void* const* d_in, const int* in_sizes, int n_in,
                              void* d_out, int out_size, void* d_ws, size_t ws_size,
                              hipStream_t stream) {
    const float* x = (const float*)d_in[0];
    const float* h = (const float*)d_in[1];
    float*       y = (float*)d_out;

    // Workspace: Wx and Wh, each ROWS * NFFT complex64 (512 MiB each, 1 GiB total).
    float2* Wx = (float2*)d_ws;
    float2* Wh = Wx + (size_t)ROWS * NFFT;

    size_t sm13 = (256 + 8 * TILE_F2) * sizeof(float2);
    size_t sm2  = (256 + 2 * 512 * 9 + 8 * TILE_F2) * sizeof(float2);

    k_fwd_inner<<<dim3(64, ROWS), 256, sm13, stream>>>(x, Wx, 1);
    k_fwd_inner<<<dim3(64, ROWS), 256, sm13, stream>>>(h, Wh, 0);
    k_outer_mul<<<dim3(32, ROWS), 256, sm2,  stream>>>(Wx, Wh);
    k_inv_inner<<<dim3(64, ROWS), 256, sm13, stream>>>(Wx, y);
}